// HiPool_58480274702406
// MI455X (gfx1250) — compile-verified
//
#include <hip/hip_runtime.h>
#include <math.h>

// ---- problem constants ----
#define S     8192
#define D     768
#define NMID  10
#define NHIGH 5
#define SEG   820     // edges_per_high at low->mid level (ceil(8192/10))
#define SEGV  205     // SEG/4
#define LSEG  812     // last cluster row count (8192 - 9*820)
#define LSEGV 203

typedef __attribute__((ext_vector_type(2))) float v2f;
typedef __attribute__((ext_vector_type(8))) float v8f;

// ---- workspace layout (float offsets) ----
static constexpr size_t OFF_ROWSEG = 0;        // 8192*10
static constexpr size_t OFF_MIDADJ = 81920;    // 100
static constexpr size_t OFF_NADJ1  = 82024;    // 100
static constexpr size_t OFF_NADJ2  = 82128;    // 25
static constexpr size_t OFF_MID    = 82160;    // 10*768
static constexpr size_t OFF_Q      = 89840;    // 16*768 (zero-padded rows 10..15)
static constexpr size_t OFF_SCORES = 102128;   // 16*8192
static constexpr size_t OFF_PART   = 233200;   // 8*16*768 (K-split partials)
static constexpr size_t OFF_REP    = 331504;   // 10*768
static constexpr size_t OFF_T1     = 339184;   // 10*768
static constexpr size_t OFF_XMID   = 346864;   // 10*768
static constexpr size_t OFF_MID2   = 354544;   // 5*768
static constexpr size_t OFF_Q2     = 358384;   // 5*768
static constexpr size_t OFF_HREP   = 362224;   // 5*768
static constexpr size_t OFF_T2     = 366064;   // 5*768

__device__ __forceinline__ float wave_sum(float v) {
  v += __shfl_xor(v, 16, 32);
  v += __shfl_xor(v,  8, 32);
  v += __shfl_xor(v,  4, 32);
  v += __shfl_xor(v,  2, 32);
  v += __shfl_xor(v,  1, 32);
  return v;
}

// K1a: per-row column-segment sums of adjacency. One block per row, 10 waves,
// wave w sums segment w with float4 loads (segments are 16B aligned).
__global__ void k_rowseg(const float* __restrict__ A, float* __restrict__ rowseg) {
  int row  = blockIdx.x;
  int w    = threadIdx.x >> 5;   // segment 0..9
  int lane = threadIdx.x & 31;
  int nvec = (w == 9) ? LSEGV : SEGV;
  const float4* p = reinterpret_cast<const float4*>(A + (size_t)row * S + (size_t)w * SEG);
  float s = 0.f;
  for (int v = lane; v < nvec; v += 32) {
    float4 t = p[v];
    s += (t.x + t.y) + (t.z + t.w);
  }
  s = wave_sum(s);
  if (lane == 0) rowseg[row * NMID + w] = s;
}

// K1b: reduce row segments over row-clusters -> mid_adj[10][10]
__global__ void k_midadj(const float* __restrict__ rowseg, float* __restrict__ midadj) {
  int idx = blockIdx.x;            // 0..99
  int ci = idx / NMID, cj = idx % NMID;
  int r0 = ci * SEG;
  int cnt = (ci == 9) ? LSEG : SEG;
  float s = 0.f;
  for (int r = threadIdx.x; r < cnt; r += 32)
    s += rowseg[(r0 + r) * NMID + cj];
  s = wave_sum(s);
  if (threadIdx.x == 0) midadj[idx] = s;
}

// Tiny: GCN-normalized adjacencies. nadj1 = norm(mid_adj with diag:=1),
// high_adj = m2^T * mid_adj(raw) * m2, nadj2 = norm(high_adj with diag:=1).
__global__ void k_adjnorm(const float* __restrict__ midadj,
                          float* __restrict__ nadj1, float* __restrict__ nadj2) {
  if (threadIdx.x != 0) return;
  float a[NMID * NMID];
  for (int i = 0; i < NMID * NMID; ++i) a[i] = midadj[i];
  float b[NMID * NMID];
  for (int i = 0; i < NMID * NMID; ++i) b[i] = a[i];
  for (int i = 0; i < NMID; ++i) b[i * NMID + i] = 1.0f;
  float dinv1[NMID];
  for (int i = 0; i < NMID; ++i) {
    float d = 0.f;
    for (int j = 0; j < NMID; ++j) d += b[i * NMID + j];
    d = fmaxf(d, 1.0f);
    dinv1[i] = 1.0f / sqrtf(d);
  }
  for (int i = 0; i < NMID; ++i)
    for (int j = 0; j < NMID; ++j)
      nadj1[i * NMID + j] = dinv1[i] * b[i * NMID + j] * dinv1[j];

  float h[NHIGH * NHIGH];
  for (int c = 0; c < NHIGH; ++c)
    for (int d = 0; d < NHIGH; ++d)
      h[c * NHIGH + d] = a[(2*c)*NMID + 2*d] + a[(2*c)*NMID + 2*d + 1]
                       + a[(2*c+1)*NMID + 2*d] + a[(2*c+1)*NMID + 2*d + 1];
  for (int c = 0; c < NHIGH; ++c) h[c * NHIGH + c] = 1.0f;
  float dinv2[NHIGH];
  for (int c = 0; c < NHIGH; ++c) {
    float d = 0.f;
    for (int j = 0; j < NHIGH; ++j) d += h[c * NHIGH + j];
    d = fmaxf(d, 1.0f);
    dinv2[c] = 1.0f / sqrtf(d);
  }
  for (int c = 0; c < NHIGH; ++c)
    for (int j = 0; j < NHIGH; ++j)
      nadj2[c * NHIGH + j] = dinv2[c] * h[c * NHIGH + j] * dinv2[j];
}

// K2: mid = m1^T x  (segment row-sums of x). grid (6, 10), block 128.
__global__ void k_mid(const float* __restrict__ x, float* __restrict__ mid) {
  int c   = blockIdx.y;
  int col = blockIdx.x * 128 + threadIdx.x;
  int r0  = c * SEG;
  int cnt = (c == 9) ? LSEG : SEG;   // both multiples of 4
  const float* p = x + (size_t)r0 * D + col;
  float s0 = 0.f, s1 = 0.f, s2 = 0.f, s3 = 0.f;
  for (int r = 0; r < cnt; r += 4) {
    s0 += p[(size_t)(r + 0) * D];
    s1 += p[(size_t)(r + 1) * D];
    s2 += p[(size_t)(r + 2) * D];
    s3 += p[(size_t)(r + 3) * D];
  }
  mid[c * D + col] = (s0 + s1) + (s2 + s3);
}

// K3: qpad = mid @ att_low_mid, zero-padded to 16 rows. grid 3, block 256.
__global__ void k_q(const float* __restrict__ mid, const float* __restrict__ att,
                    float* __restrict__ qpad) {
  __shared__ float smid[NMID * D];
  for (int i = threadIdx.x; i < NMID * D; i += 256) smid[i] = mid[i];
  __syncthreads();
  int col = blockIdx.x * 256 + threadIdx.x;
  float acc[NMID];
#pragma unroll
  for (int m = 0; m < NMID; ++m) acc[m] = 0.f;
  for (int k = 0; k < D; ++k) {
    float w = att[k * D + col];
#pragma unroll
    for (int m = 0; m < NMID; ++m) acc[m] += smid[m * D + k] * w;
  }
#pragma unroll
  for (int m = 0; m < NMID; ++m) qpad[m * D + col] = acc[m];
  for (int m = NMID; m < 16; ++m) qpad[m * D + col] = 0.f;
}

// K4 (WMMA): masked scores = (qpad @ x^T), one 16x16 output tile per wave.
// A = qpad [16 x 768], B = x rows (n0..n0+15) [768 x 16], K-loop step 4.
__global__ void k_scores(const float* __restrict__ qpad, const float* __restrict__ x,
                         float* __restrict__ scores) {
  int lane = threadIdx.x;
  int half = lane >> 4;
  int l    = lane & 15;
  int n0   = blockIdx.x * 16;
  const float* qp = qpad + l * D + 2 * half;
  const float* xp = x + (size_t)(n0 + l) * D + 2 * half;
  v8f acc = {0.f, 0.f, 0.f, 0.f, 0.f, 0.f, 0.f, 0.f};
  for (int k0 = 0; k0 < D; k0 += 4) {
    v2f a = *reinterpret_cast<const v2f*>(qp + k0);
    v2f b = *reinterpret_cast<const v2f*>(xp + k0);
    acc = __builtin_amdgcn_wmma_f32_16x16x4_f32(false, a, false, b, (short)0, acc,
                                                false, false);
  }
  int n  = n0 + l;
  int cj = n / SEG;   // own cluster of column i
#pragma unroll
  for (int v = 0; v < 8; ++v) {
    int m = v + half * 8;
    float val = acc[v];
    if (cj == m) val = 0.f;   // scores * (1 - mapping)^T : zero own-cluster entries
    scores[m * S + n] = val;
  }
}

// K5: row softmax (rows 0..9) over 8192 entries, in place.
__global__ void k_softmax(float* __restrict__ scores) {
  __shared__ float red[256];
  int m = blockIdx.x;
  float* row = scores + (size_t)m * S;
  int t = threadIdx.x;
  float mx = -1e30f;
  for (int i = t; i < S; i += 256) mx = fmaxf(mx, row[i]);
  red[t] = mx; __syncthreads();
  for (int o = 128; o > 0; o >>= 1) {
    if (t < o) red[t] = fmaxf(red[t], red[t + o]);
    __syncthreads();
  }
  mx = red[0]; __syncthreads();
  float s = 0.f;
  for (int i = t; i < S; i += 256) s += expf(row[i] - mx);
  red[t] = s; __syncthreads();
  for (int o = 128; o > 0; o >>= 1) {
    if (t < o) red[t] += red[t + o];
    __syncthreads();
  }
  float inv = 1.0f / red[0];
  for (int i = t; i < S; i += 256) row[i] = expf(row[i] - mx) * inv;
}

// K6 (WMMA): partials of scores @ x. grid (48 col-tiles, 8 K-chunks), 1 wave each.
// A = scores [16 x 8192], B = x [8192 x 768]; each block covers K range of 1024.
__global__ void k_midrep(const float* __restrict__ scores, const float* __restrict__ x,
                         float* __restrict__ part) {
  int lane = threadIdx.x;
  int half = lane >> 4;
  int l    = lane & 15;
  int n0   = blockIdx.x * 16;
  int kb   = blockIdx.y * 1024;
  const float* ap = scores + (size_t)l * S + kb + 2 * half;
  const float* bp = x + (size_t)(kb + 2 * half) * D + n0 + l;
  v8f acc = {0.f, 0.f, 0.f, 0.f, 0.f, 0.f, 0.f, 0.f};
  for (int k0 = 0; k0 < 1024; k0 += 4) {
    v2f a = *reinterpret_cast<const v2f*>(ap + k0);
    v2f b;
    b[0] = bp[(size_t)k0 * D];
    b[1] = bp[(size_t)(k0 + 1) * D];
    acc = __builtin_amdgcn_wmma_f32_16x16x4_f32(false, a, false, b, (short)0, acc,
                                                false, false);
  }
  int col = n0 + l;
#pragma unroll
  for (int v = 0; v < 8; ++v) {
    int m = v + half * 8;
    part[(size_t)(blockIdx.y * 16 + m) * D + col] = acc[v];
  }
}

// K7: deterministic reduce of K-chunk partials + residual mid -> mid_rep[10][768]
__global__ void k_repsum(const float* __restrict__ mid, const float* __restrict__ part,
                         float* __restrict__ rep) {
  int col = blockIdx.x * 256 + threadIdx.x;
  int m   = blockIdx.y;
  float s = mid[m * D + col];
#pragma unroll
  for (int kc = 0; kc < 8; ++kc) s += part[(size_t)(kc * 16 + m) * D + col];
  rep[m * D + col] = s;
}

// Generic small GEMM: out[R][768] = in[R][768] @ W[768][768]. grid 3, block 256.
template <int R>
__global__ void k_rowmm(const float* __restrict__ in, const float* __restrict__ W,
                        float* __restrict__ out) {
  __shared__ float sin_[R * D];
  for (int i = threadIdx.x; i < R * D; i += 256) sin_[i] = in[i];
  __syncthreads();
  int col = blockIdx.x * 256 + threadIdx.x;
  float acc[R];
#pragma unroll
  for (int m = 0; m < R; ++m) acc[m] = 0.f;
  for (int k = 0; k < D; ++k) {
    float w = W[k * D + col];
#pragma unroll
    for (int m = 0; m < R; ++m) acc[m] += sin_[m * D + k] * w;
  }
#pragma unroll
  for (int m = 0; m < R; ++m) out[m * D + col] = acc[m];
}

// K9: x_mid = relu(nadj1 @ t1 + b1); also mid2 = m2^T x_mid (pair sums).
__global__ void k_xmid(const float* __restrict__ nadj1, const float* __restrict__ t1,
                       const float* __restrict__ b1, float* __restrict__ xmid,
                       float* __restrict__ mid2) {
  int col = blockIdx.x * 256 + threadIdx.x;
  float t[NMID];
#pragma unroll
  for (int j = 0; j < NMID; ++j) t[j] = t1[j * D + col];
  float bias = b1[col];
  float xm[NMID];
#pragma unroll
  for (int i = 0; i < NMID; ++i) {
    float s = bias;
#pragma unroll
    for (int j = 0; j < NMID; ++j) s += nadj1[i * NMID + j] * t[j];
    xm[i] = fmaxf(s, 0.f);
    xmid[i * D + col] = xm[i];
  }
#pragma unroll
  for (int c = 0; c < NHIGH; ++c) mid2[c * D + col] = xm[2 * c] + xm[2 * c + 1];
}

// K11: level-2 attention: scores2 = q2 @ xmid^T (mask j/2==c), softmax(10),
// hrep = scores2 @ xmid + mid2. Single block.
__global__ void k_att2(const float* __restrict__ q2, const float* __restrict__ xmid,
                       const float* __restrict__ mid2, float* __restrict__ hrep) {
  __shared__ float sc[NHIGH * NMID];
  int t = threadIdx.x;
  if (t < NHIGH * NMID) {
    int c = t / NMID, j = t % NMID;
    float s = 0.f;
    for (int k = 0; k < D; ++k) s += q2[c * D + k] * xmid[j * D + k];
    if ((j >> 1) == c) s = 0.f;
    sc[t] = s;
  }
  __syncthreads();
  if (t < NHIGH) {
    float mx = -1e30f;
    for (int j = 0; j < NMID; ++j) mx = fmaxf(mx, sc[t * NMID + j]);
    float s = 0.f;
    for (int j = 0; j < NMID; ++j) s += expf(sc[t * NMID + j] - mx);
    float inv = 1.0f / s;
    for (int j = 0; j < NMID; ++j) sc[t * NMID + j] = expf(sc[t * NMID + j] - mx) * inv;
  }
  __syncthreads();
  for (int idx = t; idx < NHIGH * D; idx += blockDim.x) {
    int c = idx / D, col = idx % D;
    float s = mid2[c * D + col];
#pragma unroll
    for (int j = 0; j < NMID; ++j) s += sc[c * NMID + j] * xmid[j * D + col];
    hrep[idx] = s;
  }
}

// K13: out = mean_c relu(nadj2 @ t2 + b2)
__global__ void k_out(const float* __restrict__ nadj2, const float* __restrict__ t2,
                      const float* __restrict__ b2, float* __restrict__ out) {
  int col = blockIdx.x * 256 + threadIdx.x;
  float t[NHIGH];
#pragma unroll
  for (int m = 0; m < NHIGH; ++m) t[m] = t2[m * D + col];
  float bias = b2[col];
  float acc = 0.f;
#pragma unroll
  for (int c = 0; c < NHIGH; ++c) {
    float s = bias;
#pragma unroll
    for (int m = 0; m < NHIGH; ++m) s += nadj2[c * NHIGH + m] * t[m];
    acc += fmaxf(s, 0.f);
  }
  out[col] = acc * 0.2f;
}

extern "C" void kernel_launch(void* const* d_in, const int* in_sizes, int n_in,
                              void* d_out, int out_size, void* d_ws, size_t ws_size,
                              hipStream_t stream) {
  (void)in_sizes; (void)n_in; (void)out_size; (void)ws_size;
  const float* x    = (const float*)d_in[0];
  const float* A    = (const float*)d_in[1];
  const float* att1 = (const float*)d_in[2];
  const float* W1   = (const float*)d_in[3];
  const float* b1   = (const float*)d_in[4];
  const float* att2 = (const float*)d_in[5];
  const float* W2   = (const float*)d_in[6];
  const float* b2   = (const float*)d_in[7];
  float* out = (float*)d_out;

  float* ws = (float*)d_ws;
  float* rowseg = ws + OFF_ROWSEG;
  float* midadj = ws + OFF_MIDADJ;
  float* nadj1  = ws + OFF_NADJ1;
  float* nadj2  = ws + OFF_NADJ2;
  float* mid    = ws + OFF_MID;
  float* qpad   = ws + OFF_Q;
  float* scores = ws + OFF_SCORES;
  float* part   = ws + OFF_PART;
  float* rep    = ws + OFF_REP;
  float* t1     = ws + OFF_T1;
  float* xmid   = ws + OFF_XMID;
  float* mid2   = ws + OFF_MID2;
  float* q2     = ws + OFF_Q2;
  float* hrep   = ws + OFF_HREP;
  float* t2     = ws + OFF_T2;

  // adjacency pooling (dominant: 256 MB read)
  k_rowseg <<<S, 320, 0, stream>>>(A, rowseg);
  k_midadj <<<NMID * NMID, 32, 0, stream>>>(rowseg, midadj);
  k_adjnorm<<<1, 32, 0, stream>>>(midadj, nadj1, nadj2);

  // level-1 cluster attention
  k_mid    <<<dim3(6, 10), 128, 0, stream>>>(x, mid);
  k_q      <<<3, 256, 0, stream>>>(mid, att1, qpad);
  k_scores <<<S / 16, 32, 0, stream>>>(qpad, x, scores);
  k_softmax<<<NMID, 256, 0, stream>>>(scores);
  k_midrep <<<dim3(D / 16, 8), 32, 0, stream>>>(scores, x, part);
  k_repsum <<<dim3(3, NMID), 256, 0, stream>>>(mid, part, rep);

  // GCN 1
  k_rowmm<NMID><<<3, 256, 0, stream>>>(rep, W1, t1);
  k_xmid <<<3, 256, 0, stream>>>(nadj1, t1, b1, xmid, mid2);

  // level-2 cluster attention + GCN 2 + mean
  k_rowmm<NHIGH><<<3, 256, 0, stream>>>(mid2, att2, q2);
  k_att2 <<<1, 512, 0, stream>>>(q2, xmid, mid2, hrep);
  k_rowmm<NHIGH><<<3, 256, 0, stream>>>(hrep, W2, t2);
  k_out  <<<3, 256, 0, stream>>>(nadj2, t2, b2, out);
}